// EcsqQuan_77996606095930
// MI455X (gfx1250) — compile-verified
//
#include <hip/hip_runtime.h>
#include <math.h>
#include <stdint.h>

// ---------------------------------------------------------------------------
// CDNA5 fast tanh: v_tanh_f32 is a new gfx1250 trans op (1 instr vs ~5 for
// the exp/rcp formulation). Builtin if present, else force it via inline asm.
// ---------------------------------------------------------------------------
__device__ __forceinline__ float tanh_fast(float v) {
#if __has_builtin(__builtin_amdgcn_tanhf)
  return __builtin_amdgcn_tanhf(v);
#else
  float r;
  asm("v_tanh_f32 %0, %1" : "=v"(r) : "v"(v));
  return r;
#endif
}

// ---------------------------------------------------------------------------
// CDNA5 async global->LDS staging (ASYNCcnt-tracked, ISA ch.10 / 15.18.3)
// ---------------------------------------------------------------------------
__device__ __forceinline__ void async_load_b128(uint32_t lds_addr, const void* gaddr) {
  asm volatile("global_load_async_to_lds_b128 %0, %1, off"
               :: "v"(lds_addr), "v"(gaddr) : "memory");
}
__device__ __forceinline__ void wait_async_le2() {
#if __has_builtin(__builtin_amdgcn_s_wait_asynccnt)
  __builtin_amdgcn_s_wait_asynccnt(2);
#else
  asm volatile("s_wait_asynccnt 0x2" ::: "memory");
#endif
}
__device__ __forceinline__ void wait_async_le0() {
#if __has_builtin(__builtin_amdgcn_s_wait_asynccnt)
  __builtin_amdgcn_s_wait_asynccnt(0);
#else
  asm volatile("s_wait_asynccnt 0x0" ::: "memory");
#endif
}

// Preprocessed constants (softplus already applied to weights, tanh to factors)
struct ECConsts {
  float W0[3], W1[9], W2[9], W3[9], W4[3];
  float B0[3], B1[3], B2[3], B3[3], B4;
  float F0[3], F1[3], F2[3], F3[3];
};

// d_ws float layout
// [0]=s_c  [1]=1/s_c
// [2..4]=sp(m0) [5..13]=sp(m1) [14..22]=sp(m2) [23..31]=sp(m3) [32..34]=sp(m4)
// [35..37]=b0 [38..40]=b1 [41..43]=b2 [44..46]=b3 [47]=b4
// [48..50]=tanh(f0) [51..53]=tanh(f1) [54..56]=tanh(f2) [57..59]=tanh(f3)

__global__ void ec_precompute(
    const float* __restrict__ s,
    const float* __restrict__ m0, const float* __restrict__ b0, const float* __restrict__ f0,
    const float* __restrict__ m1, const float* __restrict__ b1, const float* __restrict__ f1,
    const float* __restrict__ m2, const float* __restrict__ b2, const float* __restrict__ f2,
    const float* __restrict__ m3, const float* __restrict__ b3, const float* __restrict__ f3,
    const float* __restrict__ m4, const float* __restrict__ b4,
    float* __restrict__ ws) {
  float sc = fmaxf(s[0], 1e-4f);
  ws[0] = sc;
  ws[1] = 1.0f / sc;
  for (int i = 0; i < 3; ++i) ws[2 + i]  = log1pf(__expf(m0[i]));
  for (int i = 0; i < 9; ++i) ws[5 + i]  = log1pf(__expf(m1[i]));
  for (int i = 0; i < 9; ++i) ws[14 + i] = log1pf(__expf(m2[i]));
  for (int i = 0; i < 9; ++i) ws[23 + i] = log1pf(__expf(m3[i]));
  for (int i = 0; i < 3; ++i) ws[32 + i] = log1pf(__expf(m4[i]));
  for (int i = 0; i < 3; ++i) ws[35 + i] = b0[i];
  for (int i = 0; i < 3; ++i) ws[38 + i] = b1[i];
  for (int i = 0; i < 3; ++i) ws[41 + i] = b2[i];
  for (int i = 0; i < 3; ++i) ws[44 + i] = b3[i];
  ws[47] = b4[0];
  for (int i = 0; i < 3; ++i) ws[48 + i] = tanhf(f0[i]);
  for (int i = 0; i < 3; ++i) ws[51 + i] = tanhf(f1[i]);
  for (int i = 0; i < 3; ++i) ws[54 + i] = tanhf(f2[i]);
  for (int i = 0; i < 3; ++i) ws[57 + i] = tanhf(f3[i]);
}

// one hidden layer: t = W*h + B ; h = t + F * tanh(t)
#define EC_LAYER(W, B, F)                                                      \
  {                                                                            \
    float t0 = fmaf(k.W[0], h0, fmaf(k.W[1], h1, fmaf(k.W[2], h2, k.B[0])));   \
    float t1 = fmaf(k.W[3], h0, fmaf(k.W[4], h1, fmaf(k.W[5], h2, k.B[1])));   \
    float t2 = fmaf(k.W[6], h0, fmaf(k.W[7], h1, fmaf(k.W[8], h2, k.B[2])));   \
    h0 = fmaf(k.F[0], tanh_fast(t0), t0);                                      \
    h1 = fmaf(k.F[1], tanh_fast(t1), t1);                                      \
    h2 = fmaf(k.F[2], tanh_fast(t2), t2);                                      \
  }

__device__ __forceinline__ float mlp_eval(float u, const ECConsts& k) {
  float t0 = fmaf(k.W0[0], u, k.B0[0]);
  float t1 = fmaf(k.W0[1], u, k.B0[1]);
  float t2 = fmaf(k.W0[2], u, k.B0[2]);
  float h0 = fmaf(k.F0[0], tanh_fast(t0), t0);
  float h1 = fmaf(k.F0[1], tanh_fast(t1), t1);
  float h2 = fmaf(k.F0[2], tanh_fast(t2), t2);
  EC_LAYER(W1, B1, F1)
  EC_LAYER(W2, B2, F2)
  EC_LAYER(W3, B3, F3)
  return fmaf(k.W4[0], h0, fmaf(k.W4[1], h1, fmaf(k.W4[2], h2, k.B4)));
}

// likelihood = max(|sigmoid(sg*up) - sigmoid(sg*lo)|, 1e-9), sg = -sign(lo+up)
// sigmoid(t) = 0.5*(1 + tanh(t/2))  ->  diff = 0.5*(tanh(sg*up/2)-tanh(sg*lo/2))
__device__ __forceinline__ float like_eval(float y, const ECConsts& k) {
  float lo = mlp_eval(y - 0.5f, k);
  float up = mlp_eval(y + 0.5f, k);
  float s  = lo + up;
  float hs = (s > 0.0f) ? -0.5f : ((s < 0.0f) ? 0.5f : 0.0f);  // sg * 0.5 folded
  float d  = tanh_fast(hs * up) - tanh_fast(hs * lo);
  return fmaxf(0.5f * fabsf(d), 1e-9f);
}

__global__ __launch_bounds__(256) void ec_fwd(
    const float* __restrict__ x, const float* __restrict__ noise,
    const float* __restrict__ cst,
    float* __restrict__ xq, float* __restrict__ lik, int n) {
  // hoist wave-uniform constants once per thread (SMEM-eligible loads)
  const float SC  = cst[0];
  const float INV = cst[1];
  ECConsts k;
#pragma unroll
  for (int i = 0; i < 3; ++i) {
    k.W0[i] = cst[2 + i];   k.W4[i] = cst[32 + i];
    k.B0[i] = cst[35 + i];  k.B1[i] = cst[38 + i];
    k.B2[i] = cst[41 + i];  k.B3[i] = cst[44 + i];
    k.F0[i] = cst[48 + i];  k.F1[i] = cst[51 + i];
    k.F2[i] = cst[54 + i];  k.F3[i] = cst[57 + i];
  }
#pragma unroll
  for (int i = 0; i < 9; ++i) {
    k.W1[i] = cst[5 + i];  k.W2[i] = cst[14 + i];  k.W3[i] = cst[23 + i];
  }
  k.B4 = cst[47];

  const int tidx = (int)threadIdx.x;

  if ((n & 3) == 0) {
    // -------- vector path: async double-buffered LDS staging --------
    const int n4 = n >> 2;
    const float4* __restrict__ x4  = (const float4*)x;
    const float4* __restrict__ nz4 = (const float4*)noise;
    float4* __restrict__ xq4 = (float4*)xq;
    float4* __restrict__ lk4 = (float4*)lik;

    __shared__ float4 sx[2][256];
    __shared__ float4 sn[2][256];

    const int nChunks = (n4 + 255) >> 8;  // 256 float4 per block-chunk

    // each thread stages exactly the float4 it will consume -> no barriers,
    // only the ASYNCcnt wait is needed (async loads complete in order)
    auto issue = [&](int chunk, int buf) {
      int idx = (chunk << 8) + tidx;
      int cl = idx < n4 ? idx : n4 - 1;  // clamp OOB lanes (dup load, unused)
      async_load_b128((uint32_t)(uintptr_t)&sx[buf][tidx], (const void*)(x4 + cl));
      async_load_b128((uint32_t)(uintptr_t)&sn[buf][tidx], (const void*)(nz4 + cl));
    };

    int c = (int)blockIdx.x;
    int buf = 0;
    if (c < nChunks) issue(c, 0);
    for (; c < nChunks; c += (int)gridDim.x, buf ^= 1) {
      const int cn = c + (int)gridDim.x;
      if (cn < nChunks) {
        issue(cn, buf ^ 1);   // prefetch next chunk while computing this one
        wait_async_le2();     // the 2 just-issued may remain outstanding
      } else {
        wait_async_le0();
      }
      const float4 xv = sx[buf][tidx];  // ds_load_b128
      const float4 nv = sn[buf][tidx];
      const int idx = (c << 8) + tidx;
      if (idx < n4) {
        float4 q;  // x_q = (x/s + noise)*s = x + noise*s
        q.x = fmaf(nv.x, SC, xv.x);
        q.y = fmaf(nv.y, SC, xv.y);
        q.z = fmaf(nv.z, SC, xv.z);
        q.w = fmaf(nv.w, SC, xv.w);
        xq4[idx] = q;

        float4 L;  // 4 independent chains -> ILP across the tanh pipe
        L.x = like_eval(xv.x * INV, k);
        L.y = like_eval(xv.y * INV, k);
        L.z = like_eval(xv.z * INV, k);
        L.w = like_eval(xv.w * INV, k);
        lk4[idx] = L;
      }
    }
  } else {
    // -------- generic scalar fallback --------
    const int tid = (int)(blockIdx.x * blockDim.x) + tidx;
    const int nthreads = (int)(gridDim.x * blockDim.x);
    for (int i = tid; i < n; i += nthreads) {
      float xv = x[i], nv = noise[i];
      xq[i]  = fmaf(nv, SC, xv);
      lik[i] = like_eval(xv * INV, k);
    }
  }
}

extern "C" void kernel_launch(void* const* d_in, const int* in_sizes, int n_in,
                              void* d_out, int out_size, void* d_ws, size_t ws_size,
                              hipStream_t stream) {
  // setup_inputs() dict order: x, noise, s, then (m_i, b_i, f_i) interleaved
  const float* x  = (const float*)d_in[0];
  const float* nz = (const float*)d_in[1];
  const float* s  = (const float*)d_in[2];
  const float* m0 = (const float*)d_in[3];
  const float* b0 = (const float*)d_in[4];
  const float* f0 = (const float*)d_in[5];
  const float* m1 = (const float*)d_in[6];
  const float* b1 = (const float*)d_in[7];
  const float* f1 = (const float*)d_in[8];
  const float* m2 = (const float*)d_in[9];
  const float* b2 = (const float*)d_in[10];
  const float* f2 = (const float*)d_in[11];
  const float* m3 = (const float*)d_in[12];
  const float* b3 = (const float*)d_in[13];
  const float* f3 = (const float*)d_in[14];
  const float* m4 = (const float*)d_in[15];
  const float* b4 = (const float*)d_in[16];

  const int n = in_sizes[0];
  float* ws  = (float*)d_ws;
  float* xq  = (float*)d_out;      // output 0: x_q  (n floats)
  float* lik = xq + n;             // output 1: likelihood (n floats)

  ec_precompute<<<1, 1, 0, stream>>>(s, m0, b0, f0, m1, b1, f1,
                                     m2, b2, f2, m3, b3, f3, m4, b4, ws);

  const int threads = 256;                    // 8 wave32 per block
  int work = ((n & 3) == 0) ? (n >> 2) : n;
  int blocks = (work + threads - 1) / threads;
  if (blocks > 2048) blocks = 2048;           // ~4 chunks per block
  if (blocks < 1) blocks = 1;
  ec_fwd<<<blocks, threads, 0, stream>>>(x, nz, ws, xq, lik, n);
}